// _Whitening_13048110646044
// MI455X (gfx1250) — compile-verified
//
#include <hip/hip_runtime.h>

#define NFEAT 256
#define GS 4
#define NG 64
#define HW 3136          // 56*56
#define NBATCH 64
#define NCOLS (NBATCH * HW)   // 200704
#define EPSV 1e-3f

typedef __attribute__((ext_vector_type(2))) float v2f;
typedef __attribute__((ext_vector_type(8))) float v8f;

// workspace layout (floats):
//  [0, 7168)     per-block stats partials: [(g*8 + blk)*14 + v]
//                v: 0..3 = channel sums, 4..13 = cov pairs
//                (0,0)(0,1)(0,2)(0,3)(1,1)(1,2)(1,3)(2,2)(2,3)(3,3)
//  [7168, 8192)  W[g][4][4] row-major (whitening matrices)
//  [8192, 8448)  bias[c] = (W_g m_g)[i]
#define WS_W    7168
#define WS_BIAS 8192

// ---------------------------------------------------------------------------
// Pass 1: per-group channel sums + raw second moments (deterministic partials)
// grid = (64 groups, 8 column-slices), block = 256. RT loads -> x fills L2.
// ---------------------------------------------------------------------------
__global__ __launch_bounds__(256) void stats_kernel(const float* __restrict__ x,
                                                    float* __restrict__ ws) {
  const int g   = blockIdx.x;
  const int blk = blockIdx.y;
  const int tid = threadIdx.x;

  float s0 = 0.f, s1 = 0.f, s2 = 0.f, s3 = 0.f;
  float c00 = 0.f, c01 = 0.f, c02 = 0.f, c03 = 0.f;
  float c11 = 0.f, c12 = 0.f, c13 = 0.f;
  float c22 = 0.f, c23 = 0.f, c33 = 0.f;

  const int base = blk * 25088;   // 200704 / 8 columns per block
  #pragma unroll 2
  for (int i = 0; i < 98; ++i) {  // 98 * 256 = 25088
    const int k  = base + i * 256 + tid;       // global column (b*HW + s)
    const int bb = k / HW;
    const int s  = k - bb * HW;
    const float* p = x + ((size_t)bb * NFEAT + g * GS) * HW + s;
    const float v0 = p[0 * HW];
    const float v1 = p[1 * HW];
    const float v2 = p[2 * HW];
    const float v3 = p[3 * HW];
    s0 += v0; s1 += v1; s2 += v2; s3 += v3;
    c00 += v0 * v0; c01 += v0 * v1; c02 += v0 * v2; c03 += v0 * v3;
    c11 += v1 * v1; c12 += v1 * v2; c13 += v1 * v3;
    c22 += v2 * v2; c23 += v2 * v3; c33 += v3 * v3;
  }

  float acc[14] = {s0, s1, s2, s3, c00, c01, c02, c03, c11, c12, c13, c22, c23, c33};

  __shared__ float lds[8][14];
  const int wv   = tid >> 5;
  const int lane = tid & 31;
  #pragma unroll
  for (int v = 0; v < 14; ++v) {
    float val = acc[v];
    #pragma unroll
    for (int off = 16; off > 0; off >>= 1) val += __shfl_down(val, off, 32);
    if (lane == 0) lds[wv][v] = val;
  }
  __syncthreads();
  if (tid < 14) {
    float s = 0.f;
    #pragma unroll
    for (int w = 0; w < 8; ++w) s += lds[w][tid];   // fixed order -> deterministic
    ws[(g * 8 + blk) * 14 + tid] = s;
  }
}

// ---------------------------------------------------------------------------
// Pass 2: per-group finalize — covariance, shrink, 4x4 Jacobi eigensolve,
// whitening matrix W = V diag(1/sqrt(lam+eps)) V^T, bias = W m.
// One thread per group.
// ---------------------------------------------------------------------------
__global__ void finalize_kernel(float* __restrict__ ws) {
  const int g = threadIdx.x;
  if (g >= NG) return;

  // deterministic sum of the 8 partials
  float acc[14];
  #pragma unroll
  for (int v = 0; v < 14; ++v) acc[v] = 0.f;
  for (int blk = 0; blk < 8; ++blk) {
    const float* p = ws + (g * 8 + blk) * 14;
    #pragma unroll
    for (int v = 0; v < 14; ++v) acc[v] += p[v];
  }

  const float n = (float)NCOLS;
  float m[4];
  #pragma unroll
  for (int i = 0; i < 4; ++i) m[i] = acc[i] / n;

  // covariance of centered data: Sxx - n*m*m^T, then shrink
  float A[4][4];
  {
    float C[4][4];
    int p = 4;
    for (int i = 0; i < 4; ++i)
      for (int j = i; j < 4; ++j) { C[i][j] = C[j][i] = acc[p++]; }
    for (int i = 0; i < 4; ++i)
      for (int j = 0; j < 4; ++j) {
        const float fc = C[i][j] - n * m[i] * m[j];
        A[i][j] = (1.f - EPSV) * fc + ((i == j) ? EPSV : 0.f);
      }
  }

  // cyclic Jacobi eigendecomposition: A = V diag(lam) V^T
  float V[4][4];
  for (int i = 0; i < 4; ++i)
    for (int j = 0; j < 4; ++j) V[i][j] = (i == j) ? 1.f : 0.f;

  const int pp[6] = {0, 0, 0, 1, 1, 2};
  const int qq[6] = {1, 2, 3, 2, 3, 3};
  for (int sweep = 0; sweep < 12; ++sweep) {
    for (int r = 0; r < 6; ++r) {
      const int p = pp[r], q = qq[r];
      const float apq = A[p][q];
      if (fabsf(apq) < 1e-30f) continue;
      const float theta = (A[q][q] - A[p][p]) / (2.f * apq);
      const float t = copysignf(1.f, theta) / (fabsf(theta) + sqrtf(theta * theta + 1.f));
      const float c = 1.f / sqrtf(t * t + 1.f);
      const float s = t * c;
      // rows p,q
      for (int k = 0; k < 4; ++k) {
        const float ap = A[p][k], aq = A[q][k];
        A[p][k] = c * ap - s * aq;
        A[q][k] = s * ap + c * aq;
      }
      // cols p,q
      for (int k = 0; k < 4; ++k) {
        const float ap = A[k][p], aq = A[k][q];
        A[k][p] = c * ap - s * aq;
        A[k][q] = s * ap + c * aq;
      }
      // accumulate eigenvectors (columns of V)
      for (int k = 0; k < 4; ++k) {
        const float vp = V[k][p], vq = V[k][q];
        V[k][p] = c * vp - s * vq;
        V[k][q] = s * vp + c * vq;
      }
    }
  }

  float rs[4];
  #pragma unroll
  for (int k = 0; k < 4; ++k) rs[k] = 1.f / sqrtf(fmaxf(A[k][k], 0.f) + EPSV);

  float* Wp = ws + WS_W + g * 16;
  float* Bp = ws + WS_BIAS + g * 4;
  float Wm[4][4];
  for (int i = 0; i < 4; ++i)
    for (int j = 0; j < 4; ++j) {
      float w = 0.f;
      #pragma unroll
      for (int k = 0; k < 4; ++k) w += V[i][k] * V[j][k] * rs[k];
      Wm[i][j] = w;
      Wp[i * 4 + j] = w;
    }
  for (int i = 0; i < 4; ++i) {
    float b = 0.f;
    #pragma unroll
    for (int j = 0; j < 4; ++j) b += Wm[i][j] * m[j];
    Bp[i] = b;
  }
}

// ---------------------------------------------------------------------------
// Pass 3: apply whitening with V_WMMA_F32_16X16X4_F32.
// Per wave: channel-block of 16 channels (4 groups) as block-diagonal A,
// 16x16 column tiles of x as B, D accumulated over four K=4 slices.
// out = W*x - bias (mean folded into bias).
// Tiles traversed in DESCENDING address order so the ~192MB tail of x left
// in L2 by the stats pass is consumed before it is evicted. x loads are
// last-use (non-temporal); out stores are non-temporal so they do not evict x.
// grid = 256 blocks x 256 threads = 2048 waves = 16 cblocks x 128 waves.
// ---------------------------------------------------------------------------
__global__ __launch_bounds__(256) void apply_kernel(const float* __restrict__ x,
                                                    float* __restrict__ out,
                                                    const float* __restrict__ ws) {
  const int tid  = threadIdx.x;
  const int lane = tid & 31;
  const int wave = (int)((blockIdx.x * blockDim.x + tid) >> 5);  // 0..2047
  const int cb   = wave >> 7;    // channel block 0..15 (16 channels each)
  const int wic  = wave & 127;   // wave index within channel block
  const int h    = lane >> 4;    // lane half
  const int nn   = lane & 15;    // N / M index

  // A operands: block-diagonal 16x16 W slice, K-split into four 16x4 pieces.
  // A layout (f32 16x4): a.x = A[M][2h], a.y = A[M][2h+1], M = lane&15.
  const float* Wp = ws + WS_W;
  const int t_of_m = nn >> 2;
  const float w0 = Wp[(cb * 4 + t_of_m) * 16 + (nn & 3) * 4 + 2 * h + 0];
  const float w1 = Wp[(cb * 4 + t_of_m) * 16 + (nn & 3) * 4 + 2 * h + 1];
  v2f a[4];
  #pragma unroll
  for (int t = 0; t < 4; ++t) {
    a[t].x = (t == t_of_m) ? w0 : 0.f;
    a[t].y = (t == t_of_m) ? w1 : 0.f;
  }

  // bias per D VGPR r: channel = cb*16 + r + 8h
  const float* Bp = ws + WS_BIAS;
  float bias[8];
  #pragma unroll
  for (int r = 0; r < 8; ++r) bias[r] = Bp[cb * 16 + r + 8 * h];

  // 12544 tiles per channel block = 64 batches * 196 spatial chunks of 16.
  // Descend: highest batch/spatial addresses first (freshest in L2).
  for (int tile = 12543 - wic; tile >= 0; tile -= 128) {
    const int bb = tile / 196;
    const int s0 = (tile - bb * 196) * 16;
    const size_t colBase = ((size_t)bb * NFEAT + (size_t)cb * 16) * HW + s0;

    // B layout (f32 4x16): b.x = B[2h][nn], b.y = B[2h+1][nn];
    // B_t row K maps to channel 4t + 2h + v.
    const float* px = x + colBase + (size_t)(2 * h) * HW + nn;
    v2f b0, b1, b2, b3;
    b0.x = __builtin_nontemporal_load(px + 0 * HW);
    b0.y = __builtin_nontemporal_load(px + 1 * HW);
    b1.x = __builtin_nontemporal_load(px + 4 * HW);
    b1.y = __builtin_nontemporal_load(px + 5 * HW);
    b2.x = __builtin_nontemporal_load(px + 8 * HW);
    b2.y = __builtin_nontemporal_load(px + 9 * HW);
    b3.x = __builtin_nontemporal_load(px + 12 * HW);
    b3.y = __builtin_nontemporal_load(px + 13 * HW);

    v8f c = {};
    c = __builtin_amdgcn_wmma_f32_16x16x4_f32(false, a[0], false, b0, (short)0, c, false, false);
    c = __builtin_amdgcn_wmma_f32_16x16x4_f32(false, a[1], false, b1, (short)0, c, false, false);
    c = __builtin_amdgcn_wmma_f32_16x16x4_f32(false, a[2], false, b2, (short)0, c, false, false);
    c = __builtin_amdgcn_wmma_f32_16x16x4_f32(false, a[3], false, b3, (short)0, c, false, false);

    // D layout: VGPR r -> channel cb*16 + r + 8h, column s0 + nn
    float* po = out + colBase + (size_t)(8 * h) * HW + nn;
    #pragma unroll
    for (int r = 0; r < 8; ++r)
      __builtin_nontemporal_store(c[r] - bias[r], po + (size_t)r * HW);
  }
}

// ---------------------------------------------------------------------------
extern "C" void kernel_launch(void* const* d_in, const int* in_sizes, int n_in,
                              void* d_out, int out_size, void* d_ws, size_t ws_size,
                              hipStream_t stream) {
  (void)in_sizes; (void)n_in; (void)out_size; (void)ws_size;
  const float* x = (const float*)d_in[0];
  float* out = (float*)d_out;
  float* ws  = (float*)d_ws;

  stats_kernel<<<dim3(NG, 8), 256, 0, stream>>>(x, ws);
  finalize_kernel<<<1, 64, 0, stream>>>(ws);
  apply_kernel<<<256, 256, 0, stream>>>(x, out, ws);
}